// GNN_27539330302469
// MI455X (gfx1250) — compile-verified
//
#include <hip/hip_runtime.h>
#include <hip/hip_bf16.h>

typedef __attribute__((ext_vector_type(2))) float v2f;
typedef __attribute__((ext_vector_type(8))) float v8f;

// ---------------------------------------------------------------- utilities
__global__ void k_zero_i32(int* __restrict__ p, int n) {
    int i = blockIdx.x * blockDim.x + threadIdx.x;
    if (i < n) p[i] = 0;
}

__global__ void k_copy_i32(const int* __restrict__ a, int* __restrict__ b, int n) {
    int i = blockIdx.x * blockDim.x + threadIdx.x;
    if (i < n) b[i] = a[i];
}

// in-degree counts (int atomics; only E of them, trivial)
__global__ void k_cnt(const long long* __restrict__ dst, int* __restrict__ cnt, long long E) {
    long long i = (long long)blockIdx.x * blockDim.x + threadIdx.x;
    if (i < E) atomicAdd(&cnt[dst[i]], 1);
}

// dinv = rsqrt(2 + in_degree)   (improved=True self-loop weight 2)
__global__ void k_dinv(const int* __restrict__ cnt, float* __restrict__ dinv, int n) {
    int i = blockIdx.x * blockDim.x + threadIdx.x;
    if (i < n) dinv[i] = rsqrtf(2.0f + (float)cnt[i]);
}

// exclusive scan of cnt[0..n) into rowptr[0..n]; single 1024-thread workgroup
__global__ void k_scan(const int* __restrict__ cnt, int* __restrict__ rowptr, int n) {
    __shared__ int smem[1024];
    __shared__ int carry;
    if (threadIdx.x == 0) carry = 0;
    __syncthreads();
    for (int base = 0; base < n; base += 1024) {
        int i = base + (int)threadIdx.x;
        int v = (i < n) ? cnt[i] : 0;
        smem[threadIdx.x] = v;
        __syncthreads();
        #pragma unroll
        for (int off = 1; off < 1024; off <<= 1) {
            int t = (threadIdx.x >= off) ? smem[threadIdx.x - off] : 0;
            __syncthreads();
            smem[threadIdx.x] += t;
            __syncthreads();
        }
        int incl = smem[threadIdx.x];
        if (i < n) rowptr[i] = carry + (incl - v);
        __syncthreads();
        if (threadIdx.x == 1023) carry += incl;   // incl of last thread = chunk total
        __syncthreads();
    }
    if (threadIdx.x == 0) rowptr[n] = carry;
}

// bucket edges by destination: srcIdx/coef in dst-sorted order
__global__ void k_fillcsr(const long long* __restrict__ src, const long long* __restrict__ dst,
                          const float* __restrict__ dinv, int* __restrict__ cursor,
                          int* __restrict__ srcIdx, float* __restrict__ coef, long long E) {
    long long e = (long long)blockIdx.x * blockDim.x + threadIdx.x;
    if (e >= E) return;
    long long s = src[e], d = dst[e];
    int pos = atomicAdd(&cursor[d], 1);
    srcIdx[pos] = (int)s;
    coef[pos] = dinv[s] * dinv[d];
}

// dst[rd x cd] = src[rs x cs] zero-padded
__global__ void k_pad(const float* __restrict__ src, float* __restrict__ dst,
                      int rs, int cs, int rd, int cd) {
    int i = blockIdx.x * blockDim.x + threadIdx.x;
    if (i >= rd * cd) return;
    int r = i / cd, c = i % cd;
    dst[i] = (r < rs && c < cs) ? src[r * cs + c] : 0.0f;
}

// ---------------------------------------------------------------- CSR gather
// one block per destination node, one thread per feature: register
// accumulation over in-edges, coalesced row reads, NO atomics, no pre-zero.
__global__ void k_gather(const int* __restrict__ rowptr, const int* __restrict__ srcIdx,
                         const float* __restrict__ coef, const float* __restrict__ Hin,
                         float* __restrict__ Agg, int C) {
    int n = blockIdx.x;
    int t = threadIdx.x;
    int beg = rowptr[n], end = rowptr[n + 1];
    float acc = 0.0f;
    for (int j = beg; j < end; ++j) {
        long long s = srcIdx[j];
        if (j + 1 < end) {
            long long sn = srcIdx[j + 1];
            __builtin_prefetch(&Hin[sn * (long long)C + t], 0, 3);
        }
        acc += Hin[s * (long long)C + t] * coef[j];
    }
    Agg[(long long)n * C + t] = acc;
}

// Agg = relu(Agg + (2*dinv^2)*H1 + b1)   (in place)
__global__ void k_relu(const float* __restrict__ H1, float* __restrict__ Agg,
                       const float* __restrict__ dinv, const float* __restrict__ b1,
                       int C, long long total) {
    long long i = (long long)blockIdx.x * blockDim.x + threadIdx.x;
    if (i >= total) return;
    long long n = i / C;
    int c = (int)(i % C);
    float di = dinv[n];
    float v = Agg[i] + (2.0f * di * di) * H1[i] + b1[c];
    Agg[i] = v > 0.0f ? v : 0.0f;
}

__device__ __forceinline__ float softplus_f(float x) {
    return fmaxf(x, 0.0f) + log1pf(expf(-fabsf(x)));
}

// epilogue of layer 2: mr = softplus(Agg2 + self*H2 + b2) -> d_out,
// H2 <- tanh(softplus(thr)*mr)  (padded cols zeroed)
__global__ void k_epi(const float* __restrict__ Agg2, float* __restrict__ H2,
                      const float* __restrict__ dinv, const float* __restrict__ b2,
                      const float* __restrict__ thr, float* __restrict__ mr_out,
                      int Cpad, int Cval, long long total) {
    long long i = (long long)blockIdx.x * blockDim.x + threadIdx.x;
    if (i >= total) return;
    long long n = i / Cpad;
    int c = (int)(i % Cpad);
    if (c < Cval) {
        float di = dinv[n];
        float v = Agg2[i] + (2.0f * di * di) * H2[i] + b2[c];
        float sp = softplus_f(v);
        mr_out[n * (long long)Cval + c] = sp;
        float k = softplus_f(thr[0]);
        H2[i] = tanhf(k * sp);
    } else {
        H2[i] = 0.0f;
    }
}

// ---------------------------------------------------------------- WMMA GEMM
// C[M x Npad] = A[M x K] @ B[K x Npad] (+bias). One wave computes a 32x32
// tile via 2x2 register blocking: 2 A frags x 2 B frags -> 4
// V_WMMA_F32_16X16X4_F32 per K-step (8 FLOP/byte from cache vs 4 unblocked).
// M and Npad must be multiples of 32. Stores only cols < nvalid.
__global__ void k_gemm_wmma(const float* __restrict__ A, const float* __restrict__ B,
                            float* __restrict__ C, const float* __restrict__ bias,
                            int K, int lda, int ldb, int ldc,
                            int tilesN2, int nvalid, int totalTiles) {
    int wave = blockIdx.x * (blockDim.x >> 5) + (threadIdx.x >> 5);
    if (wave >= totalTiles) return;            // wave-uniform exit: EXEC stays all-ones
    int lane = threadIdx.x & 31;
    int tm = wave / tilesN2;                   // 32-row supertile
    int tn = wave % tilesN2;                   // 32-col supertile

    // A layout (16x4 f32): lanes 0-15 & 16-31 both map M=0..15; K pair differs
    int khi   = (lane >> 4) << 1;              // 0 (lanes 0-15) or 2 (lanes 16-31)
    int bcol0 = tn * 32 + (lane & 15);
    int bcol1 = bcol0 + 16;

    const float* A0 = A + (long long)(tm * 32 + (lane & 15)) * lda;
    const float* A1 = A0 + (long long)16 * lda;

    v8f acc00 = {}, acc01 = {}, acc10 = {}, acc11 = {};
    #pragma unroll 2
    for (int k0 = 0; k0 < K; k0 += 4) {
        const float* Bk = B + (long long)(k0 + khi) * ldb;
        // prefetch next K-panel (speculative; dropped harmlessly past the end)
        __builtin_prefetch(A0 + k0 + 16, 0, 3);
        __builtin_prefetch(Bk + (long long)4 * ldb + bcol0, 0, 3);

        v2f a0, a1, b0, b1;
        a0.x = A0[k0 + khi];     a0.y = A0[k0 + khi + 1];
        a1.x = A1[k0 + khi];     a1.y = A1[k0 + khi + 1];
        b0.x = Bk[bcol0];        b0.y = Bk[ldb + bcol0];
        b1.x = Bk[bcol1];        b1.y = Bk[ldb + bcol1];

        acc00 = __builtin_amdgcn_wmma_f32_16x16x4_f32(false, a0, false, b0, (short)0, acc00, false, false);
        acc01 = __builtin_amdgcn_wmma_f32_16x16x4_f32(false, a0, false, b1, (short)0, acc01, false, false);
        acc10 = __builtin_amdgcn_wmma_f32_16x16x4_f32(false, a1, false, b0, (short)0, acc10, false, false);
        acc11 = __builtin_amdgcn_wmma_f32_16x16x4_f32(false, a1, false, b1, (short)0, acc11, false, false);
    }

    // C/D layout: lanes 0-15 hold M=r, lanes 16-31 hold M=r+8 (r = VGPR 0..7)
    int rsub = (lane >> 4) << 3;
    int cg0  = tn * 32 + (lane & 15);
    int cg1  = cg0 + 16;
    float bad0 = (bias && cg0 < nvalid) ? bias[cg0] : 0.0f;
    float bad1 = (bias && cg1 < nvalid) ? bias[cg1] : 0.0f;
    #pragma unroll
    for (int r = 0; r < 8; ++r) {
        long long row0 = (long long)(tm * 32 + rsub + r) * ldc;
        long long row1 = (long long)(tm * 32 + 16 + rsub + r) * ldc;
        if (cg0 < nvalid) {
            C[row0 + cg0] = acc00[r] + bad0;
            C[row1 + cg0] = acc10[r] + bad0;
        }
        if (cg1 < nvalid) {
            C[row0 + cg1] = acc01[r] + bad1;
            C[row1 + cg1] = acc11[r] + bad1;
        }
    }
}

// ---------------------------------------------------------------- launch
static inline unsigned cdiv(long long a, long long b) { return (unsigned)((a + b - 1) / b); }

extern "C" void kernel_launch(void* const* d_in, const int* in_sizes, int n_in,
                              void* d_out, int out_size, void* d_ws, size_t ws_size,
                              hipStream_t stream) {
    const float*     x    = (const float*)d_in[0];
    const long long* ei   = (const long long*)d_in[1];
    const float*     W1   = (const float*)d_in[2];
    const float*     b1   = (const float*)d_in[3];
    const float*     W2   = (const float*)d_in[4];
    const float*     b2   = (const float*)d_in[5];
    const float*     Wout = (const float*)d_in[6];
    const float*     bout = (const float*)d_in[7];
    const float*     thr  = (const float*)d_in[8];

    const int DIN = 128, Hc = 256, Mc = 500, Mp = 512, Oc = 100, Op = 128;
    const int N = in_sizes[0] / DIN;               // 100000 (divisible by 32)
    const long long E = (long long)in_sizes[1] / 2;
    const long long* src = ei;
    const long long* dst = ei + E;

    // ---------------- workspace layout (floats unless noted)
    float* ws    = (float*)d_ws;
    float* H1    = ws;                              // [N,256]
    float* AGG1  = ws + (size_t)N * Hc;             // [N,256]; AGG2 = H1..AGG1 as [N,512]
    float* AGG2  = H1;
    float* H2    = ws + (size_t)N * Mp;             // [N,512]
    float* dinv  = ws + (size_t)N * 1024;           // [N]
    float* W2p   = dinv + N;                        // [256,512]
    float* Woutp = W2p + (size_t)Hc * Mp;           // [512,128]
    float* coef  = Woutp + (size_t)Mp * Op;         // [E]
    int*   ibase = (int*)(coef + E);
    int*   cnt    = ibase;                          // [N]
    int*   rowptr = ibase + N;                      // [N+1]
    int*   cursor = rowptr + (N + 1);               // [N]
    int*   srcIdx = cursor + N;                     // [E]

    float* out_mr = (float*)d_out;                  // [N,500]
    float* out_y  = (float*)d_out + (size_t)N * Mc; // [N,100]

    // ---------------- CSR build (counts -> scan -> bucket fill) + dinv
    k_zero_i32<<<cdiv(N, 256), 256, 0, stream>>>(cnt, N);
    k_cnt     <<<cdiv(E, 256), 256, 0, stream>>>(dst, cnt, E);
    k_dinv    <<<cdiv(N, 256), 256, 0, stream>>>(cnt, dinv, N);
    k_scan    <<<1, 1024, 0, stream>>>(cnt, rowptr, N);
    k_copy_i32<<<cdiv(N, 256), 256, 0, stream>>>(rowptr, cursor, N);
    k_fillcsr <<<cdiv(E, 256), 256, 0, stream>>>(src, dst, dinv, cursor, srcIdx, coef, E);

    // padded weights (zero pad rows/cols so padded p/H2 columns are inert)
    k_pad<<<cdiv((long long)Hc * Mp, 256), 256, 0, stream>>>(W2,   W2p,   Hc, Mc, Hc, Mp);
    k_pad<<<cdiv((long long)Mp * Op, 256), 256, 0, stream>>>(Wout, Woutp, Mc, Oc, Mp, Op);

    // layer 1: H1 = x @ W1; AGG1 = gather(H1); relu+self+bias in place
    {
        int tiles = (N / 32) * (Hc / 32);
        k_gemm_wmma<<<cdiv(tiles, 4), 128, 0, stream>>>(x, W1, H1, nullptr,
                                                        DIN, DIN, Hc, Hc, Hc / 32, Hc, tiles);
    }
    k_gather<<<dim3((unsigned)N), Hc, 0, stream>>>(rowptr, srcIdx, coef, H1, AGG1, Hc);
    k_relu<<<cdiv((long long)N * Hc, 256), 256, 0, stream>>>(H1, AGG1, dinv, b1, Hc,
                                                             (long long)N * Hc);

    // layer 2: H2 = relu(h) @ W2p; AGG2 = gather(H2); fused softplus/tanh epilogue
    {
        int tiles = (N / 32) * (Mp / 32);
        k_gemm_wmma<<<cdiv(tiles, 4), 128, 0, stream>>>(AGG1, W2p, H2, nullptr,
                                                        Hc, Hc, Mp, Mp, Mp / 32, Mp, tiles);
    }
    k_gather<<<dim3((unsigned)N), Mp, 0, stream>>>(rowptr, srcIdx, coef, H2, AGG2, Mp);
    k_epi<<<cdiv((long long)N * Mp, 256), 256, 0, stream>>>(AGG2, H2, dinv, b2, thr, out_mr,
                                                            Mp, Mc, (long long)N * Mp);

    // head: y_go = p @ Woutp + bout  (store cols < 100)
    {
        int tiles = (N / 32) * (Op / 32);
        k_gemm_wmma<<<cdiv(tiles, 4), 128, 0, stream>>>(H2, Woutp, out_y, bout,
                                                        Mp, Mp, Op, Oc, Op / 32, Oc, tiles);
    }
}